// TransformerBlock_75600014344435
// MI455X (gfx1250) — compile-verified
//
#include <hip/hip_runtime.h>
#include <math.h>

// ---------------------------------------------------------------------------
// GPT-OSS-style transformer block on gfx1250 (MI455X), all-fp32 via
// V_WMMA_F32_16X16X4_F32.  B=1, S=256, H=512, 8 heads x d64 (GQA 2 KV),
// sliding window 128 + attention sinks, 128-expert top-2 MoE (swiglu).
// ---------------------------------------------------------------------------

typedef __attribute__((ext_vector_type(2))) float v2f;
typedef __attribute__((ext_vector_type(8))) float v8f;

__device__ __forceinline__ v8f wmma_f32(v2f a, v2f b, v8f c) {
  // D(16x16,f32) = A(16x4,f32) * B(4x16,f32) + C
  return __builtin_amdgcn_wmma_f32_16x16x4_f32(
      /*neg_a=*/false, a, /*neg_b=*/false, b,
      /*c_mod=*/(short)0, c, /*reuse_a=*/false, /*reuse_b=*/false);
}

static constexpr int SEQ  = 256;
static constexpr int HDIM = 512;
static constexpr int QKVD = 768;   // 8*64 q + 2*64 k + 2*64 v
static constexpr int NEXP = 128;

// --------------------------- RMSNorm ---------------------------------------
__global__ void k_rmsnorm(const float* __restrict__ x, const float* __restrict__ w,
                          float* __restrict__ y) {
  int tok = blockIdx.x;
  int tid = threadIdx.x;                      // 256 threads
  const float* xr = x + tok * HDIM;
  float s = 0.f;
  for (int i = tid; i < HDIM; i += 256) { float v = xr[i]; s += v * v; }
  __shared__ float red[256];
  red[tid] = s; __syncthreads();
  for (int off = 128; off > 0; off >>= 1) {
    if (tid < off) red[tid] += red[tid + off];
    __syncthreads();
  }
  float inv = rsqrtf(red[0] * (1.f / HDIM) + 1e-5f);
  for (int i = tid; i < HDIM; i += 256) y[tok * HDIM + i] = xr[i] * inv * w[i];
}

// ------------------ generic GEMM, W stored [K][N] row-major -----------------
// C[M][N] = A[M][K] * W[K][N]  (+ optional residual).  One wave per 16x16 tile.
__global__ void k_gemm_kn(const float* __restrict__ A, const float* __restrict__ W,
                          float* __restrict__ C, const float* __restrict__ resid,
                          int M, int N, int K) {
  int wave   = threadIdx.x >> 5;
  int tile   = blockIdx.x * (blockDim.x >> 5) + wave;
  int ntiles = (M >> 4) * (N >> 4);
  if (tile >= ntiles) return;                 // wave-uniform
  int nT = N >> 4;
  int mb = (tile / nT) << 4;
  int nb = (tile % nT) << 4;
  int lane = threadIdx.x & 31;
  int r  = lane & 15;
  int kh = lane >> 4;
  const float* arow = A + (size_t)(mb + r) * K + 2 * kh;
  const float* wcol = W + nb + r + (size_t)(2 * kh) * N;
  v8f acc = {0.f, 0.f, 0.f, 0.f, 0.f, 0.f, 0.f, 0.f};
  for (int k0 = 0; k0 < K; k0 += 4) {
    float2 av = *(const float2*)(arow + k0);
    v2f a = {av.x, av.y};
    v2f b = {wcol[(size_t)k0 * N], wcol[(size_t)k0 * N + N]};
    acc = wmma_f32(a, b, acc);
  }
#pragma unroll
  for (int v = 0; v < 8; ++v) {
    int row = mb + v + 8 * kh;
    int idx = row * N + nb + r;
    C[idx] = resid ? (acc[v] + resid[idx]) : acc[v];
  }
}

// ------------------------------- YaRN RoPE ----------------------------------
__global__ void k_rope(float* __restrict__ qkv) {
  int tok  = blockIdx.x;
  int tid  = threadIdx.x;                     // 320 = 10 heads * 32 pairs
  int head = tid >> 5;                        // 0..7 = q heads, 8..9 = k heads
  int i    = tid & 31;
  int off  = (head < 8) ? head * 64 : 512 + (head - 8) * 64;
  float* p = qkv + tok * QKVD + off;

  const float TWO_PI = 6.28318530717958647692f;
  float fi     = (float)i;
  float freq   = __powf(150000.f, fi * (1.f / 32.f));
  float interp = 1.f / (32.f * freq);
  float extrap = 1.f / freq;
  float lb     = __logf(150000.f);
  float low    = 32.f * __logf(4096.f / (32.f * TWO_PI)) / lb;
  float high   = 32.f * __logf(4096.f / TWO_PI) / lb;
  float ramp   = fminf(fmaxf((fi - low) / (high - low), 0.f), 1.f);
  float mask   = 1.f - ramp;
  float invf   = interp * (1.f - mask) + extrap * mask;
  float conc   = 0.1f * __logf(32.f) + 1.f;
  float ang    = (float)tok * invf;
  float c = __cosf(ang) * conc, s = __sinf(ang) * conc;
  float x1 = p[i], x2 = p[i + 32];
  p[i]      = x1 * c - x2 * s;
  p[i + 32] = x2 * c + x1 * s;
}

// --------------------- attention: one wave per (head, 16 queries) -----------
__global__ void k_attn(const float* __restrict__ qkv, const float* __restrict__ sinks,
                       float* __restrict__ attn) {
  int h  = blockIdx.x;                        // 0..7
  int qb = blockIdx.y;                        // 0..15
  int g  = h >> 2;                            // KV group
  int q0 = qb << 4;
  int kst    = (q0 > 127) ? ((q0 - 127) >> 4) : 0;
  int ntiles = (q0 >> 4) - kst + 1;           // <= 9
  int L      = ntiles << 4;

  __shared__ float sc[16][160];
  int lane = threadIdx.x & 31;
  int r  = lane & 15;
  int kh = lane >> 4;

  // ---- scores: Q(16x64) * K^T ----
  const float* qrow = qkv + (q0 + r) * QKVD + h * 64 + 2 * kh;
  for (int t = 0; t < ntiles; ++t) {
    int kb = (kst + t) << 4;
    const float* krow = qkv + (kb + r) * QKVD + 512 + g * 64 + 2 * kh;
    v8f acc = {0.f, 0.f, 0.f, 0.f, 0.f, 0.f, 0.f, 0.f};
    for (int k0 = 0; k0 < 64; k0 += 4) {
      float2 av = *(const float2*)(qrow + k0);
      float2 bv = *(const float2*)(krow + k0);
      v2f a = {av.x, av.y};
      v2f b = {bv.x, bv.y};
      acc = wmma_f32(a, b, acc);
    }
#pragma unroll
    for (int v = 0; v < 8; ++v) {
      int m  = v + 8 * kh;
      int q  = q0 + m;
      int kk = kb + r;
      bool ok = (kk <= q) && (kk + 128 > q);  // causal AND sliding window
      sc[m][t * 16 + r] = ok ? acc[v] * 0.125f : -INFINITY;
    }
  }
  __syncthreads();

  // ---- softmax with sink column (sink contributes to denominator only) ----
  if (lane < 16) {
    float snk = sinks[h];
    float mx = snk;
    for (int j = 0; j < L; ++j) mx = fmaxf(mx, sc[lane][j]);
    float d = __expf(snk - mx);
    for (int j = 0; j < L; ++j) { float p = __expf(sc[lane][j] - mx); sc[lane][j] = p; d += p; }
    float inv = 1.f / d;
    for (int j = 0; j < L; ++j) sc[lane][j] *= inv;
  }
  __syncthreads();

  // ---- P(16xL) * V(Lx64) ----
  v8f o[4];
#pragma unroll
  for (int t = 0; t < 4; ++t) o[t] = (v8f){0.f, 0.f, 0.f, 0.f, 0.f, 0.f, 0.f, 0.f};
  int kbase = kst << 4;
  for (int k0 = 0; k0 < L; k0 += 4) {
    float2 av = *(const float2*)(&sc[r][k0 + 2 * kh]);
    v2f a = {av.x, av.y};
    const float* vrow = qkv + (kbase + k0 + 2 * kh) * QKVD + 640 + g * 64 + r;
#pragma unroll
    for (int t = 0; t < 4; ++t) {
      v2f b = {vrow[t * 16], vrow[QKVD + t * 16]};
      o[t] = wmma_f32(a, b, o[t]);
    }
  }
#pragma unroll
  for (int v = 0; v < 8; ++v) {
    int row = q0 + v + 8 * kh;
#pragma unroll
    for (int t = 0; t < 4; ++t)
      attn[row * HDIM + h * 64 + t * 16 + r] = o[t][v];
  }
}

// ------------------------------ routing -------------------------------------
__global__ void k_route(const float* __restrict__ g, int* __restrict__ ei,
                        float* __restrict__ ew) {
  int tok = blockIdx.x * blockDim.x + threadIdx.x;
  if (tok >= SEQ) return;
  const float* gr = g + tok * NEXP;
  float b1 = -INFINITY, b2 = -INFINITY;
  int i1 = 0, i2 = 0;
  for (int j = 0; j < NEXP; ++j) {
    float v = gr[j];
    if (v > b1) { b2 = b1; i2 = i1; b1 = v; i1 = j; }
    else if (v > b2) { b2 = v; i2 = j; }
  }
  float e1 = __expf(b2 - b1);
  float s = 1.f + e1;
  ei[tok * 2] = i1; ei[tok * 2 + 1] = i2;
  ew[tok * 2] = 1.f / s; ew[tok * 2 + 1] = e1 / s;
}

__global__ void k_zero(float* __restrict__ acc, int* __restrict__ cnt) {
  int i = blockIdx.x * 256 + threadIdx.x;
  if (i < SEQ * HDIM) acc[i] = 0.f;
  if (i < NEXP) cnt[i] = 0;
}

__global__ void k_scatter(const int* __restrict__ ei, const float* __restrict__ ew,
                          int* __restrict__ cnt, int* __restrict__ toks,
                          float* __restrict__ wts) {
  int t = blockIdx.x * blockDim.x + threadIdx.x;
  if (t >= SEQ * 2) return;
  int e = ei[t];
  int pos = atomicAdd(&cnt[e], 1);
  toks[e * 512 + pos] = t >> 1;
  wts[e * 512 + pos]  = ew[t];
}

// ------------------ MoE: one workgroup per expert, 16-token WMMA tiles ------
__global__ void __launch_bounds__(256)
k_moe(const float* __restrict__ tn, const float* __restrict__ w1,
      const float* __restrict__ b1, const float* __restrict__ w2,
      const float* __restrict__ b2, const int* __restrict__ cnt,
      const int* __restrict__ toks, const float* __restrict__ wts,
      float* __restrict__ acc) {
  int e = blockIdx.x;
  int n = cnt[e];
  if (n == 0) return;
  const float* W1 = w1 + (size_t)e * 512 * 512;   // [512 out][512 in]
  const float* B1 = b1 + e * 512;
  const float* W2 = w2 + (size_t)e * 512 * 256;   // [512 out][256 in]
  const float* B2 = b2 + e * 512;

  __shared__ float t1[16][512];                   // mlp1 pre-activations (32KB)
  __shared__ float hb[16][256];                   // swiglu outputs       (16KB)
  __shared__ int   ltok[16];
  __shared__ float lwt[16];

  int tid  = threadIdx.x;                         // 256 = 8 waves
  int wave = tid >> 5;
  int lane = tid & 31;
  int r  = lane & 15;
  int kh = lane >> 4;

  for (int t0 = 0; t0 < n; t0 += 16) {
    if (tid < 16) {
      int idx = t0 + tid;
      ltok[tid] = toks[e * 512 + ((idx < n) ? idx : t0)];  // clamp pads
      lwt[tid]  = (idx < n) ? wts[e * 512 + idx] : 0.f;
    }
    __syncthreads();

    const float* arow = tn + (size_t)ltok[r] * 512 + 2 * kh;

    // ---- mlp1: 512 outputs -> 32 n-tiles, 4 per wave ----
#pragma unroll
    for (int j = 0; j < 4; ++j) {
      int nb = (wave * 4 + j) << 4;
      const float* wr = W1 + (size_t)(nb + r) * 512 + 2 * kh;
      v8f a8 = {0.f, 0.f, 0.f, 0.f, 0.f, 0.f, 0.f, 0.f};
      for (int k0 = 0; k0 < 512; k0 += 4) {
        float2 av = *(const float2*)(arow + k0);
        float2 bv = *(const float2*)(wr + k0);
        v2f a = {av.x, av.y};
        v2f b = {bv.x, bv.y};
        a8 = wmma_f32(a, b, a8);
      }
      float bias = B1[nb + r];
#pragma unroll
      for (int v = 0; v < 8; ++v) t1[v + 8 * kh][nb + r] = a8[v] + bias;
    }
    __syncthreads();

    // ---- swiglu (interleaved gate/linear pairs) ----
    for (int idx = tid; idx < 16 * 256; idx += 256) {
      int m = idx >> 8, c = idx & 255;
      float xg = fminf(t1[m][2 * c], 7.f);
      float xl = fminf(fmaxf(t1[m][2 * c + 1], -7.f), 7.f);
      float sig = 1.f / (1.f + __expf(-1.702f * xg));
      hb[m][c] = xg * sig * (xl + 1.f);
    }
    __syncthreads();

    // ---- mlp2: 512 outputs -> 32 n-tiles, 4 per wave; weighted atomic out ----
#pragma unroll
    for (int j = 0; j < 4; ++j) {
      int nb = (wave * 4 + j) << 4;
      const float* wr = W2 + (size_t)(nb + r) * 256 + 2 * kh;
      v8f a8 = {0.f, 0.f, 0.f, 0.f, 0.f, 0.f, 0.f, 0.f};
      for (int k0 = 0; k0 < 256; k0 += 4) {
        float2 av = *(const float2*)(&hb[r][k0 + 2 * kh]);
        float2 bv = *(const float2*)(wr + k0);
        v2f a = {av.x, av.y};
        v2f b = {bv.x, bv.y};
        a8 = wmma_f32(a, b, a8);
      }
      float bias = B2[nb + r];
#pragma unroll
      for (int v = 0; v < 8; ++v) {
        int m = v + 8 * kh;
        if (t0 + m < n)
          atomicAdd(&acc[(size_t)ltok[m] * 512 + nb + r], (a8[v] + bias) * lwt[m]);
      }
    }
    __syncthreads();
  }
}

__global__ void k_addout(const float* __restrict__ x2, const float* __restrict__ moe,
                         float* __restrict__ out) {
  int i = blockIdx.x * 256 + threadIdx.x;
  if (i < SEQ * HDIM) out[i] = x2[i] + moe[i];
}

// ---------------------------------------------------------------------------
extern "C" void kernel_launch(void* const* d_in, const int* in_sizes, int n_in,
                              void* d_out, int out_size, void* d_ws, size_t ws_size,
                              hipStream_t stream) {
  const float* x      = (const float*)d_in[0];
  const float* n1w    = (const float*)d_in[1];
  const float* qkv_w  = (const float*)d_in[2];
  const float* out_w  = (const float*)d_in[3];
  const float* sinks  = (const float*)d_in[4];
  const float* n2w    = (const float*)d_in[5];
  const float* gate_w = (const float*)d_in[6];
  const float* m1w    = (const float*)d_in[7];
  const float* m1b    = (const float*)d_in[8];
  const float* m2w    = (const float*)d_in[9];
  const float* m2b    = (const float*)d_in[10];
  float* out = (float*)d_out;

  char* w = (char*)d_ws;
  auto alloc = [&](size_t bytes) {
    void* p = (void*)w;
    w += (bytes + 255) & ~(size_t)255;
    return p;
  };
  float* tn   = (float*)alloc(SEQ * HDIM * 4);
  float* qkv  = (float*)alloc(SEQ * QKVD * 4);
  float* attn = (float*)alloc(SEQ * HDIM * 4);
  float* x2   = (float*)alloc(SEQ * HDIM * 4);
  float* t2n  = (float*)alloc(SEQ * HDIM * 4);
  float* g    = (float*)alloc(SEQ * NEXP * 4);
  int*   ei   = (int*)alloc(SEQ * 2 * 4);
  float* ew   = (float*)alloc(SEQ * 2 * 4);
  int*   cnt  = (int*)alloc(NEXP * 4);
  int*   toks = (int*)alloc(NEXP * 512 * 4);
  float* wts  = (float*)alloc(NEXP * 512 * 4);
  float* macc = (float*)alloc(SEQ * HDIM * 4);

  k_rmsnorm<<<SEQ, 256, 0, stream>>>(x, n1w, tn);
  // qkv: 256x768x512 -> 16*48 = 768 tiles, 8 waves/block
  k_gemm_kn<<<96, 256, 0, stream>>>(tn, qkv_w, qkv, nullptr, SEQ, QKVD, HDIM);
  k_rope<<<SEQ, 320, 0, stream>>>(qkv);
  dim3 ag(8, 16);
  k_attn<<<ag, 32, 0, stream>>>(qkv, sinks, attn);
  // out proj + residual: 16*32 = 512 tiles
  k_gemm_kn<<<64, 256, 0, stream>>>(attn, out_w, x2, x, SEQ, HDIM, HDIM);
  k_rmsnorm<<<SEQ, 256, 0, stream>>>(x2, n2w, t2n);
  // router: 16*8 = 128 tiles
  k_gemm_kn<<<16, 256, 0, stream>>>(t2n, gate_w, g, nullptr, SEQ, NEXP, HDIM);
  k_route<<<1, 256, 0, stream>>>(g, ei, ew);
  k_zero<<<512, 256, 0, stream>>>(macc, cnt);
  k_scatter<<<2, 256, 0, stream>>>(ei, ew, cnt, toks, wts);
  k_moe<<<NEXP, 256, 0, stream>>>(t2n, m1w, m1b, m2w, m2b, cnt, toks, wts, macc);
  k_addout<<<512, 256, 0, stream>>>(x2, macc, out);
}